// QAConvLoss_43104291783357
// MI455X (gfx1250) — compile-verified
//
#include <hip/hip_runtime.h>
#include <hip/hip_bf16.h>
#include <math.h>

typedef __attribute__((ext_vector_type(2))) float v2f;
typedef __attribute__((ext_vector_type(8))) float v8f;

#define NCLS 1000
#define CH   128
#define HWSZ 96          // H*W = 12*8
#define BB   16
#define STR  130         // padded LDS row stride (floats), even for float2 loads
#define TILE_F (HWSZ*STR)   // 12480 floats per staged matrix

// monotonic float<->uint encoding for atomic max (handles negatives)
__device__ inline unsigned f2ord(float f) {
    unsigned u = __float_as_uint(f);
    return (u & 0x80000000u) ? ~u : (u | 0x80000000u);
}
__device__ inline float ord2f(unsigned u) {
    unsigned v = (u & 0x80000000u) ? (u & 0x7FFFFFFFu) : ~u;
    return __uint_as_float(v);
}

__device__ inline float log_sigmoid(float z) {
    return (z >= 0.f) ? -log1pf(expf(-z)) : (z - log1pf(expf(z)));
}

// ---------------------------------------------------------------------------
// Kernel 1: one block per class. Computes, for all 16 batches, the 96x96
// correlation matrix via V_WMMA_F32_16X16X4_F32 (two independent accumulator
// chains over K for ILP), reduces row/col maxima, and emits per-(m,b):
// dot(score, fc_w), sum(score), sum(score^2).
// ---------------------------------------------------------------------------
__global__ __launch_bounds__(256)
void k_pairscore(const float* __restrict__ feature,      // [B,C,H,W]
                 const float* __restrict__ cmem,         // [N,C,H,W]
                 const float* __restrict__ fc_w,         // [192]
                 float* __restrict__ dots,               // [N*B]
                 float* __restrict__ partS,              // [N]
                 float* __restrict__ partQ)              // [N]
{
    extern __shared__ float lds[];
    float*    smem_mem = lds;                       // 12480
    float*    sfea     = smem_mem + TILE_F;         // 2*12480 (two batches)
    float*    minv     = sfea + 2*TILE_F;           // 96
    float*    finv     = minv + HWSZ;               // 192
    float*    fcw_s    = finv + 2*HWSZ;             // 192
    float*    redS     = fcw_s + 192;               // 256
    float*    redQ     = redS + 256;                // 256
    float*    redD     = redQ + 256;                // 256
    unsigned* maxU     = (unsigned*)(redD + 256);   // 2*192

    const int m    = blockIdx.x;
    const int tid  = threadIdx.x;
    const int wave = tid >> 5;
    const int lane = tid & 31;
    const int lr   = lane & 15;   // row within 16 (M for A, N for B)
    const int hi   = lane >> 4;   // selects K-pair {0,1} vs {2,3}

    // ---- stage class tile, transposed to [q][c] with padded stride ----
    for (int i = tid; i < CH*HWSZ; i += 256) {
        int c = i / HWSZ, q = i % HWSZ;
        smem_mem[q*STR + c] = cmem[(size_t)m*CH*HWSZ + c*HWSZ + q];
    }
    for (int i = tid; i < 2*192; i += 256) maxU[i] = 0u;
    if (tid < 192) fcw_s[tid] = fc_w[tid];
    __syncthreads();
    for (int q = tid; q < HWSZ; q += 256) {
        float ss = 0.f;
        for (int c = 0; c < CH; ++c) { float v = smem_mem[q*STR + c]; ss += v*v; }
        minv[q] = 1.0f / fmaxf(sqrtf(ss), 1e-12f);
    }
    __syncthreads();
    for (int i = tid; i < CH*HWSZ; i += 256) {
        int q = i / CH, c = i % CH;
        smem_mem[q*STR + c] *= minv[q];
    }
    __syncthreads();

    float accS = 0.f, accQ = 0.f;   // only thread 0's copies are used

    for (int bp = 0; bp < BB/2; ++bp) {
        // ---- stage + normalize two batches of features: [p][c] ----
        for (int i = tid; i < 2*CH*HWSZ; i += 256) {
            int half = i / (CH*HWSZ);
            int r = i % (CH*HWSZ);
            int c = r / HWSZ, p = r % HWSZ;
            sfea[half*TILE_F + p*STR + c] =
                feature[(size_t)(2*bp + half)*CH*HWSZ + c*HWSZ + p];
        }
        __syncthreads();
        for (int t = tid; t < 2*HWSZ; t += 256) {
            int half = t / HWSZ, p = t % HWSZ;
            float ss = 0.f;
            for (int c = 0; c < CH; ++c) { float v = sfea[half*TILE_F + p*STR + c]; ss += v*v; }
            finv[t] = 1.0f / fmaxf(sqrtf(ss), 1e-12f);
        }
        __syncthreads();
        for (int i = tid; i < 2*CH*HWSZ; i += 256) {
            int half = i / (CH*HWSZ);
            int r = i % (CH*HWSZ);
            int p = r / CH, c = r % CH;
            sfea[half*TILE_F + p*STR + c] *= finv[half*HWSZ + p];
        }
        __syncthreads();

        // ---- WMMA tile jobs: waves 0-3 -> batch 2bp, waves 4-7 -> 2bp+1 ----
        const int half = wave >> 2;
        for (int job = (wave & 3); job < 36; job += 4) {
            const int pt = job / 6, qt = job % 6;
            const float* Abase = &sfea[half*TILE_F + (pt*16 + lr)*STR + 2*hi];
            const float* Bbase = &smem_mem[(qt*16 + lr)*STR + 2*hi];
            // two independent accumulation chains over K (even/odd 4-wide steps)
            v8f acc0 = {0.f,0.f,0.f,0.f,0.f,0.f,0.f,0.f};
            v8f acc1 = {0.f,0.f,0.f,0.f,0.f,0.f,0.f,0.f};
            #pragma unroll
            for (int k = 0; k < 16; ++k) {
                v2f a0 = *reinterpret_cast<const v2f*>(Abase + 8*k);
                v2f b0 = *reinterpret_cast<const v2f*>(Bbase + 8*k);
                v2f a1 = *reinterpret_cast<const v2f*>(Abase + 8*k + 4);
                v2f b1 = *reinterpret_cast<const v2f*>(Bbase + 8*k + 4);
                acc0 = __builtin_amdgcn_wmma_f32_16x16x4_f32(
                    false, a0, false, b0, (short)0, acc0, false, false);
                acc1 = __builtin_amdgcn_wmma_f32_16x16x4_f32(
                    false, a1, false, b1, (short)0, acc1, false, false);
            }
            v8f acc;
            #pragma unroll
            for (int r = 0; r < 8; ++r) acc[r] = acc0[r] + acc1[r];

            // column max (over M within tile): per-lane max over 8 regs,
            // lanes l and l+16 both hit slot qt*16+lr via atomic max.
            float cm = acc[0];
            #pragma unroll
            for (int r = 1; r < 8; ++r) cm = fmaxf(cm, acc[r]);
            atomicMax(&maxU[half*192 + qt*16 + lr], f2ord(cm));
            // row max (over N): butterfly across the 16-lane half-groups
            #pragma unroll
            for (int r = 0; r < 8; ++r) {
                float v = acc[r];
                v = fmaxf(v, __shfl_xor(v, 1, 32));
                v = fmaxf(v, __shfl_xor(v, 2, 32));
                v = fmaxf(v, __shfl_xor(v, 4, 32));
                v = fmaxf(v, __shfl_xor(v, 8, 32));
                if (lr == 0)
                    atomicMax(&maxU[half*192 + 96 + pt*16 + hi*8 + r], f2ord(v));
            }
        }
        __syncthreads();

        // ---- epilogue: per (m,b) deterministic tree reductions ----
        for (int h = 0; h < 2; ++h) {
            float s = 0.f, w = 0.f;
            if (tid < 192) { s = ord2f(maxU[h*192 + tid]); w = fcw_s[tid]; }
            redS[tid] = (tid < 192) ? s     : 0.f;
            redQ[tid] = (tid < 192) ? s*s   : 0.f;
            redD[tid] = (tid < 192) ? s*w   : 0.f;
            __syncthreads();
            for (int st = 128; st > 0; st >>= 1) {
                if (tid < st) {
                    redS[tid] += redS[tid+st];
                    redQ[tid] += redQ[tid+st];
                    redD[tid] += redD[tid+st];
                }
                __syncthreads();
            }
            if (tid == 0) {
                accS += redS[0];
                accQ += redQ[0];
                dots[m*BB + 2*bp + h] = redD[0];
            }
            __syncthreads();
        }
        // reset maxima for next pair
        for (int i = tid; i < 2*192; i += 256) maxU[i] = 0u;
        __syncthreads();
    }

    if (tid == 0) { partS[m] = accS; partQ[m] = accQ; }
}

// ---------------------------------------------------------------------------
// Kernel 2: BN1 (collapsed into affine on dot), logits, BN2, focal BCE + acc.
// ---------------------------------------------------------------------------
__global__ __launch_bounds__(1024)
void k_final(const float* __restrict__ dots, float* __restrict__ logits,
             const float* __restrict__ partS, const float* __restrict__ partQ,
             const float* __restrict__ fc_w, const float* __restrict__ fc_b,
             const float* __restrict__ bn_w, const float* __restrict__ bn_b,
             const float* __restrict__ lbn_w, const float* __restrict__ lbn_b,
             const int* __restrict__ target, float* __restrict__ out)
{
    __shared__ double rA[1024], rB[1024];
    __shared__ double sd[6];
    __shared__ float lossA[1024], maxV[1024];
    __shared__ int   maxI[1024];
    const int t = threadIdx.x;

    // global score stats from per-class partials
    rA[t] = (t < NCLS) ? (double)partS[t] : 0.0;
    rB[t] = (t < NCLS) ? (double)partQ[t] : 0.0;
    __syncthreads();
    for (int st = 512; st > 0; st >>= 1) {
        if (t < st) { rA[t] += rA[t+st]; rB[t] += rB[t+st]; }
        __syncthreads();
    }
    if (t == 0) { sd[0] = rA[0]; sd[1] = rB[0]; }
    __syncthreads();
    // Wsum = sum(fc_w)
    rA[t] = (t < 192) ? (double)fc_w[t] : 0.0;
    __syncthreads();
    for (int st = 512; st > 0; st >>= 1) {
        if (t < st) rA[t] += rA[t+st];
        __syncthreads();
    }
    if (t == 0) {
        const double cnt = (double)NCLS * BB * 192.0;
        double mean = sd[0] / cnt;
        double var  = sd[1] / cnt - mean*mean;
        double inv  = 1.0 / sqrt(var + 1e-5);
        double alpha = inv * (double)bn_w[0];
        double beta  = ((double)bn_b[0] - mean*alpha) * rA[0] + (double)fc_b[0];
        sd[2] = alpha; sd[3] = beta;
    }
    __syncthreads();

    // logits + BN2 stats
    const float alpha = (float)sd[2], beta = (float)sd[3];
    double s2 = 0.0, q2 = 0.0;
    for (int i = t; i < NCLS*BB; i += 1024) {
        float L = alpha * dots[i] + beta;
        logits[i] = L;
        s2 += (double)L;
        q2 += (double)L * (double)L;
    }
    rA[t] = s2; rB[t] = q2;
    __syncthreads();
    for (int st = 512; st > 0; st >>= 1) {
        if (t < st) { rA[t] += rA[t+st]; rB[t] += rB[t+st]; }
        __syncthreads();
    }
    if (t == 0) {
        const double cnt = (double)NCLS * BB;
        double m2 = rA[0] / cnt;
        double v2 = rB[0] / cnt - m2*m2;
        double inv2 = 1.0 / sqrt(v2 + 1e-5);
        double la = inv2 * (double)lbn_w[0];
        sd[4] = la; sd[5] = (double)lbn_b[0] - m2*la;
    }
    __syncthreads();

    // focal BCE + argmax: 16 groups of 64 threads, one batch each
    const float la = (float)sd[4], lb = (float)sd[5];
    const int g = t >> 6, ln = t & 63;
    const int tgt = target[g];
    float loss = 0.f, bestV = -INFINITY;
    int bestI = 0x7fffffff;
    for (int n = ln; n < NCLS; n += 64) {
        float z = la * logits[n*BB + g] + lb;
        if (z > bestV) { bestV = z; bestI = n; }
        bool oh = (n == tgt);
        float bce  = oh ? -log_sigmoid(z) : -log_sigmoid(-z);
        float prob = 1.f / (1.f + expf(-z));
        float wgt  = oh ? (1.f - prob) : prob;
        loss += bce * wgt * wgt;
    }
    lossA[t] = loss; maxV[t] = bestV; maxI[t] = bestI;
    __syncthreads();
    for (int st = 32; st > 0; st >>= 1) {
        if (ln < st) {
            int o = t + st;
            lossA[t] += lossA[o];
            if (maxV[o] > maxV[t] || (maxV[o] == maxV[t] && maxI[o] < maxI[t])) {
                maxV[t] = maxV[o]; maxI[t] = maxI[o];
            }
        }
        __syncthreads();
    }
    if (ln == 0) {
        out[g]      = lossA[t];
        out[16 + g] = (maxI[t] == tgt) ? 1.f : 0.f;
    }
}

extern "C" void kernel_launch(void* const* d_in, const int* in_sizes, int n_in,
                              void* d_out, int out_size, void* d_ws, size_t ws_size,
                              hipStream_t stream) {
    const float* feature = (const float*)d_in[0];
    const int*   target  = (const int*)  d_in[1];
    const float* cmem    = (const float*)d_in[2];
    const float* bn_w    = (const float*)d_in[3];
    const float* bn_b    = (const float*)d_in[4];
    const float* fc_w    = (const float*)d_in[5];
    const float* fc_b    = (const float*)d_in[6];
    const float* lbn_w   = (const float*)d_in[7];
    const float* lbn_b   = (const float*)d_in[8];
    float* out = (float*)d_out;

    float* ws     = (float*)d_ws;
    float* dots   = ws;                 // 16000
    float* logits = ws + 16000;         // 16000
    float* pS     = ws + 32000;         // 1000
    float* pQ     = ws + 33000;         // 1000

    const size_t smem = (size_t)(TILE_F*3 + HWSZ + 2*HWSZ + 192 + 3*256 + 2*192) * 4;

    k_pairscore<<<NCLS, 256, smem, stream>>>(feature, cmem, fc_w, dots, pS, pQ);
    k_final<<<1, 1024, 0, stream>>>(dots, logits, pS, pQ, fc_w, fc_b,
                                    bn_w, bn_b, lbn_w, lbn_b, target, out);
}